// PairUpdate_17609365914012
// MI455X (gfx1250) — compile-verified
//
#include <hip/hip_runtime.h>

typedef float v2f __attribute__((ext_vector_type(2)));
typedef float v4f __attribute__((ext_vector_type(4)));
typedef float v8f __attribute__((ext_vector_type(8)));

#define LSEQ 768
#define DSGL 384
#define DPAIR 128

// ---------------------------------------------------------------------------
// Stage 1: row = s @ W_a^T + b ; col = s @ W_b^T   (each 768x128, K=384)
// One wave per 16x16 output tile, V_WMMA_F32_16X16X4_F32, 96 K-steps.
// A lane (r = lane%16, h = lane/16) holds A[r][2h],A[r][2h+1] and
// B[2h][r],B[2h+1][r]; with B[k][n] = W[p0+n][d0+k] both operands are
// contiguous float2 loads from row-major s and W.
// ---------------------------------------------------------------------------
__global__ __launch_bounds__(256) void pair_gemm_kernel(
    const float* __restrict__ s, const float* __restrict__ W,
    const float* __restrict__ b, float* __restrict__ rowb,
    float* __restrict__ colb) {
  const int wave = (blockIdx.x * 256 + threadIdx.x) >> 5; // 0..767
  const int lane = threadIdx.x & 31;
  const int mat  = wave / 384;       // 0 -> row (W_a), 1 -> col (W_b)
  const int rem  = wave % 384;
  const int l0   = (rem >> 3) * 16;  // 48 l-tiles
  const int p0   = (rem & 7) * 16;   // 8 p-tiles
  const int r    = lane & 15;
  const int h    = lane >> 4;

  const float* sA = s + (size_t)(l0 + r) * DSGL + 2 * h;
  const float* sB = W + (size_t)(p0 + r) * (2 * DSGL) + mat * DSGL + 2 * h;

  v8f acc = {};
#pragma unroll 4
  for (int d0 = 0; d0 < DSGL; d0 += 4) {
    v2f a  = *(const v2f*)(sA + d0);
    v2f bb = *(const v2f*)(sB + d0);
    acc = __builtin_amdgcn_wmma_f32_16x16x4_f32(
        /*neg_a=*/false, a, /*neg_b=*/false, bb,
        /*c_mod=*/(short)0, acc, /*reuse_a=*/false, /*reuse_b=*/false);
  }

  const float bias = (mat == 0) ? b[p0 + r] : 0.0f; // fold bias into row table
  float* dst = (mat == 0) ? rowb : colb;
#pragma unroll
  for (int v = 0; v < 8; ++v) {
    // D layout: VGPR v holds D[M=8h+v][N=r]
    dst[(size_t)(l0 + 8 * h + v) * DPAIR + (p0 + r)] = acc[v] + bias;
  }
}

// ---------------------------------------------------------------------------
// Stage 2: out[i,j,p] = z[i,j,p] + rowb[i,p] + colb[j,p]
// Pure HBM-bandwidth stage (604 MB @ 23.3 TB/s). float4 loads/stores,
// non-temporal hints on the touch-once z/out streams, row[i] in LDS,
// col[j] served from L2 (393 KB table, L2 = 192 MB).
// Block: 256 threads = 8 j-values x 32 float4 of p; 16 j per block.
// ---------------------------------------------------------------------------
__global__ __launch_bounds__(256) void pair_add_kernel(
    const v4f* __restrict__ z, const float* __restrict__ rowb,
    const float* __restrict__ colb, v4f* __restrict__ out) {
  __shared__ v4f srow[32];
  const int i = blockIdx.y;
  if (threadIdx.x < 32)
    srow[threadIdx.x] = ((const v4f*)(rowb + (size_t)i * DPAIR))[threadIdx.x];
  __syncthreads();

  const int jj    = threadIdx.x >> 5;  // 0..7
  const int p4    = threadIdx.x & 31;  // float4 index within 128-wide p
  const v4f rv    = srow[p4];
  const int jbase = blockIdx.x * 16;

#pragma unroll
  for (int it = 0; it < 2; ++it) {
    const int    j   = jbase + it * 8 + jj;
    const size_t idx = ((size_t)i * LSEQ + j) * (DPAIR / 4) + p4;
    v4f cv = *(const v4f*)(colb + (size_t)j * DPAIR + p4 * 4); // L2-hot
    v4f zc = __builtin_nontemporal_load(z + idx);
    v4f o  = zc + rv + cv;
    __builtin_nontemporal_store(o, out + idx);
  }
}

extern "C" void kernel_launch(void* const* d_in, const int* in_sizes, int n_in,
                              void* d_out, int out_size, void* d_ws,
                              size_t ws_size, hipStream_t stream) {
  const float* s = (const float*)d_in[0]; // (768, 384)
  const float* z = (const float*)d_in[1]; // (768, 768, 128)
  const float* W = (const float*)d_in[2]; // (128, 768)
  const float* b = (const float*)d_in[3]; // (128,)
  float* out = (float*)d_out;

  float* rowb = (float*)d_ws;            // 768*128 f32 (row + bias)
  float* colb = rowb + LSEQ * DPAIR;     // 768*128 f32

  // 768 tiles total (2 mats x 48 l-tiles x 8 p-tiles), 8 waves/block
  pair_gemm_kernel<<<96, 256, 0, stream>>>(s, W, b, rowb, colb);

  dim3 grid(LSEQ / 16, LSEQ);
  pair_add_kernel<<<grid, 256, 0, stream>>>((const v4f*)z, rowb, colb,
                                            (v4f*)out);
}